// CauDilConv1D_57312043598389
// MI455X (gfx1250) — compile-verified
//
#include <hip/hip_runtime.h>

typedef __attribute__((ext_vector_type(2))) float v2f;
typedef __attribute__((ext_vector_type(4))) float v4f;
typedef __attribute__((ext_vector_type(8))) float v8f;

#define NB    16      // batch
#define CIN   512
#define COUT  512
#define TLEN  8192
#define KW    4
#define DIL   8
#define PADL  24      // (KW-1)*DIL

#define TM    128     // out-channel tile per workgroup
#define TN    128     // time tile per workgroup
#define CB    16      // input channels per K chunk
#define NCH   (CIN / CB)    // 32 chunks
#define XT    (TN + PADL)   // 152 floats of x per channel row (38 float4)
#define XP    156           // padded LDS row stride (floats, 16B-aligned)

#if defined(__has_builtin)
#  if __has_builtin(__builtin_amdgcn_global_load_async_to_lds_b128) && \
      __has_builtin(__builtin_amdgcn_s_wait_asynccnt)
#    define USE_ASYNC 1
#  endif
#endif
#ifndef USE_ASYNC
#  define USE_ASYNC 0
#endif

#if USE_ASYNC
#pragma message("CDNA5 async global->LDS path ENABLED")
// Builtin signature (from hipcc diagnostic): parameters are
//   (int __vector_size(16) __device__ *src, int __vector_size(16) __shared__ *dst,
//    imm int offset, imm int cpol)
typedef int v4i __attribute__((vector_size(16)));
typedef __attribute__((address_space(1))) v4i* g_v4i_ptr;   // pointer to AS1 int4
typedef __attribute__((address_space(3))) v4i* l_v4i_ptr;   // pointer to AS3 int4
#else
#pragma message("CDNA5 async global->LDS path DISABLED (fallback sync staging)")
#endif

// Stage one chunk-row slice: copies x float4s j4 in [6,38) always, and
// j4 in [0,6) only when t0>0 (t0==0 halo is pre-zeroed once in the prologue).
// xrow/ldsrow are per-lane base pointers at float4 index lj of the row.
__device__ __forceinline__ void stage_chunk(const float* xrow, float* ldsrow,
                                            int t0, int lj)
{
#if USE_ASYNC
    if (t0 > 0 && lj < 6)
        __builtin_amdgcn_global_load_async_to_lds_b128(
            (g_v4i_ptr)xrow, (l_v4i_ptr)ldsrow, 0, 0);
    __builtin_amdgcn_global_load_async_to_lds_b128(
        (g_v4i_ptr)xrow, (l_v4i_ptr)ldsrow, 96, 0);   // j4 = 6 + lj
    __builtin_amdgcn_global_load_async_to_lds_b128(
        (g_v4i_ptr)xrow, (l_v4i_ptr)ldsrow, 352, 0);  // j4 = 22 + lj
#else
    if (t0 > 0 && lj < 6)
        *(v4f*)ldsrow = *(const v4f*)xrow;
    *(v4f*)(ldsrow + 24) = *(const v4f*)(xrow + 24);  // 96 B
    *(v4f*)(ldsrow + 88) = *(const v4f*)(xrow + 88);  // 352 B
#endif
}

__device__ __forceinline__ void stage_wait()
{
#if USE_ASYNC
    __builtin_amdgcn_s_wait_asynccnt(0);
#endif
}

__global__ __launch_bounds__(256)
void CauDilConv1D_wmma_f32(const float* __restrict__ x,
                           const float* __restrict__ W,
                           float* __restrict__ out)
{
    __shared__ float lds_x[2][CB * XP];   // 2 * 16 * 156 * 4 = 19968 bytes

    const int tid  = threadIdx.x;
    const int lane = tid & 31;
    const int wave = tid >> 5;         // 0..7
    const int wm   = wave & 3;         // wave M row (0..3)
    const int wn   = wave >> 2;        // wave N col (0..1)
    const int l16  = lane & 15;
    const int lhi  = lane >> 4;        // 0 for lanes 0-15, 1 for 16-31

    const int t0 = blockIdx.x * TN;    // time tile origin
    const int o0 = blockIdx.y * TM;    // out-channel tile origin
    const int b  = blockIdx.z;

    const float* xb = x + (size_t)b * CIN * TLEN;

    // A-operand (W) per-lane row bases for the wave's two 16-row M subtiles.
    // fp32 WMMA A layout: lane m holds K=0,1 (VGPR0,1); lane m+16 holds K=2,3.
    // W[o, c*4+k] is contiguous in (c,k) -> per-lane float2 at W[row, 4c + 2*lhi].
    const float* wrow0 = W + (size_t)(o0 + wm * 32 +      l16) * (CIN * KW) + 2 * lhi;
    const float* wrow1 = W + (size_t)(o0 + wm * 32 + 16 + l16) * (CIN * KW) + 2 * lhi;

    v8f acc[2][4];
    #pragma unroll
    for (int sm = 0; sm < 2; ++sm)
        #pragma unroll
        for (int n = 0; n < 4; ++n)
            acc[sm][n] = (v8f){0.f, 0.f, 0.f, 0.f, 0.f, 0.f, 0.f, 0.f};

    // x staging mapping: 16 threads per channel row, float4 transfers
    const int lc = tid >> 4;           // channel row 0..15
    const int lj = tid & 15;           // float4 lane within row

    // Per-lane base pointers (row origin corresponds to t = t0 - 24; the
    // pointer is only dereferenced at byte offsets >= 96 when t0 == 0).
    const float* xstage  = xb + (size_t)lc * TLEN + (t0 - PADL) + 4 * lj;
    float*       lstage0 = &lds_x[0][lc * XP + 4 * lj];
    float*       lstage1 = &lds_x[1][lc * XP + 4 * lj];

    // Causal halo: first 24 floats of every row are zero for all chunks when
    // t0 == 0. Fill both buffers once; async loads never overwrite this range.
    if (t0 == 0 && tid < 192) {
        int bufi = tid / 96;           // buffer 0/1
        int rr   = (tid % 96) / 6;     // row 0..15
        int qq   = tid % 6;            // float4 0..5
        *(v4f*)&lds_x[bufi][rr * XP + 4 * qq] = (v4f){0.f, 0.f, 0.f, 0.f};
    }

    // Prologue: stage chunk 0 into buffer 0.
    stage_chunk(xstage, lstage0, t0, lj);
    stage_wait();
    __syncthreads();

    for (int ch = 0; ch < NCH; ++ch) {
        // Issue next chunk's async loads into the other buffer. Safe: the
        // barrier ending iteration ch-1 guarantees all waves finished
        // reading that buffer.
        if (ch + 1 < NCH)
            stage_chunk(xstage + (size_t)(ch + 1) * CB * TLEN,
                        ((ch + 1) & 1) ? lstage1 : lstage0, t0, lj);

        const float* xtile = lds_x[ch & 1];
        const int c0 = ch * CB;

        #pragma unroll 4
        for (int cc = 0; cc < CB; ++cc) {
            const int c = c0 + cc;
            // A tiles: W[o0+wm*32(+16)+m, 4c .. 4c+3] split across lane halves
            v2f a0 = *(const v2f*)(wrow0 + 4 * c);
            v2f a1 = *(const v2f*)(wrow1 + 4 * c);
            const float* xl = xtile + cc * XP;
            #pragma unroll
            for (int n = 0; n < 4; ++n) {
                // B tile (4x16): VGPR0 -> K = lhi, VGPR1 -> K = 2 + lhi.
                // im2col row c*4+k reads x at local time j = n_local + 8*k.
                int j0 = wn * 64 + n * 16 + l16 + 8 * lhi;
                v2f bv;
                bv.x = xl[j0];         // k = lhi
                bv.y = xl[j0 + 16];    // k = lhi + 2
                acc[0][n] = __builtin_amdgcn_wmma_f32_16x16x4_f32(
                    false, a0, false, bv, (short)0, acc[0][n], false, false);
                acc[1][n] = __builtin_amdgcn_wmma_f32_16x16x4_f32(
                    false, a1, false, bv, (short)0, acc[1][n], false, false);
            }
        }

        stage_wait();       // this wave's async writes for chunk ch+1 landed
        __syncthreads();    // all waves' writes landed -> safe to read
    }

    // Store: C/D layout -> VGPR r, lane half lhi: M = r + 8*lhi, N = l16.
    float* ob = out + (size_t)b * COUT * TLEN;
    #pragma unroll
    for (int sm = 0; sm < 2; ++sm) {
        #pragma unroll
        for (int n = 0; n < 4; ++n) {
            float* op = ob + (size_t)(o0 + wm * 32 + sm * 16 + 8 * lhi) * TLEN
                           + (t0 + wn * 64 + n * 16 + l16);
            #pragma unroll
            for (int r = 0; r < 8; ++r)
                op[(size_t)r * TLEN] = acc[sm][n][r];
        }
    }
}

extern "C" void kernel_launch(void* const* d_in, const int* in_sizes, int n_in,
                              void* d_out, int out_size, void* d_ws, size_t ws_size,
                              hipStream_t stream) {
    const float* x = (const float*)d_in[0];   // [16, 512, 8192] f32
    const float* W = (const float*)d_in[1];   // [512, 2048]     f32
    float* out = (float*)d_out;               // [16, 512, 8192] f32

    dim3 grid(TLEN / TN, COUT / TM, NB);      // 64 x 4 x 16
    dim3 block(256);                          // 8 waves (wave32)
    hipLaunchKernelGGL(CauDilConv1D_wmma_f32, grid, block, 0, stream, x, W, out);
}